// EEGNET_VIT_A_78477642433029
// MI455X (gfx1250) — compile-verified
//
#include <hip/hip_runtime.h>
#include <math.h>

// ---------------------------------------------------------------------------
// EEGNet-ViT fused inference for gfx1250 (MI455X), wave32.
// One wave (32 threads) per batch element; all intermediates in LDS.
// Stage A (channel-collapse GEMM 16x64x125) uses V_WMMA_F32_16X16X4_F32.
// Stage B (63-tap conv) uses halo-padded LDS + uniform-broadcast weights.
// ---------------------------------------------------------------------------

#define BATCH 2048
#define CEEG  64
#define TLEN  125
#define EPS   1e-5f

typedef float v2f __attribute__((ext_vector_type(2)));
typedef float v8f __attribute__((ext_vector_type(8)));

struct EegParams {
    const float *x, *conv_t_w, *conv_t_b, *bn1_g, *bn1_b, *bn1_m, *bn1_v,
                *conv_s_w, *conv_s_b, *bn2_g, *bn2_b, *bn2_m, *bn2_v,
                *proj_w, *proj_b, *cls_token, *pe,
                *qkv_w, *qkv_b, *out_w, *out_b, *ln1_g, *ln1_b, *ln2_g, *ln2_b,
                *mlp_w1, *mlp_b1, *mlp_w2, *mlp_b2,
                *head_ln_g, *head_ln_b, *head_w, *head_b;
    float* out;
};

// N-length dot product, both sides 16B aligned -> b128 loads.
template <int N>
__device__ __forceinline__ float dotN(const float* __restrict__ w,
                                      const float* __restrict__ h) {
    const float4* w4 = (const float4*)w;
    const float4* h4 = (const float4*)h;
    float acc = 0.f;
#pragma unroll
    for (int q = 0; q < N / 4; ++q) {
        float4 a = w4[q], bb = h4[q];
        acc += a.x * bb.x + a.y * bb.y + a.z * bb.z + a.w * bb.w;
    }
    return acc;
}

__global__ __launch_bounds__(32)
void eeg_vit_fused(EegParams P) {
    const int b    = blockIdx.x;
    const int lane = threadIdx.x;
    const int m    = lane & 15;
    const int hi   = lane >> 4;

    // halo-padded channel-collapsed signal: valid t at j = t + 32, j in [32,157)
    __shared__ alignas(16) float upad[16][192];
    __shared__ alignas(16) float ebuf[16][128];  // post conv_t+BN2+ELU (t<125)
    __shared__ alignas(16) float wtf[8][64];     // BN1-folded temporal weights
    __shared__ float A2s[16], B2s[16];
    __shared__ alignas(16) float pool[16][25];
    __shared__ alignas(16) float zs[6][16];      // transformer state
    __shared__ alignas(16) float h1[6][16], h2[6][16];
    __shared__ alignas(16) float qkvs[6][192];
    __shared__ alignas(16) float attn_o[6][64];
    __shared__ alignas(16) float m1[6][32];

    // ---- zero the conv halos: j in [0,32) and [160,192)
    for (int idx = lane; idx < 16 * 64; idx += 32) {
        int o = idx >> 6, jj = idx & 63;
        upad[o][(jj < 32) ? jj : (128 + jj)] = 0.f;
    }
    // ---- fold BN1 into temporal weights: wtf[f][k] = a1_f * conv_t_w[f][k]
    for (int idx = lane; idx < 8 * 63; idx += 32) {
        int f = idx / 63;
        float a1 = P.bn1_g[f] * rsqrtf(P.bn1_v[f] + EPS);
        wtf[f][idx % 63] = a1 * P.conv_t_w[idx];
    }
    // ---- per-output-channel affine (conv biases + BN1 const + BN2 folded)
    if (lane < 16) {
        int o = lane, f = o >> 1;
        float a1 = P.bn1_g[f] * rsqrtf(P.bn1_v[f] + EPS);
        float c1 = (P.conv_t_b[f] - P.bn1_m[f]) * a1 + P.bn1_b[f];
        float S = 0.f;
        for (int c = 0; c < CEEG; ++c) S += P.conv_s_w[o * CEEG + c];
        float d  = c1 * S + P.conv_s_b[o];
        float a2 = P.bn2_g[o] * rsqrtf(P.bn2_v[o] + EPS);
        A2s[o] = a2;
        B2s[o] = a2 * (d - P.bn2_m[o]) + P.bn2_b[o];
    }

    // ================= Stage A: u = Ws(16x64) @ X(64x125) via WMMA f32 =====
    // A layout (16x4 f32): lanes 0-15 hold K={0,1}, lanes 16-31 K={2,3}, M=lane%16
    v2f afr[16];
#pragma unroll
    for (int ks = 0; ks < 16; ++ks) {
        int c0 = 4 * ks + 2 * hi;
        afr[ks].x = P.conv_s_w[m * CEEG + c0];
        afr[ks].y = P.conv_s_w[m * CEEG + c0 + 1];
    }
    const float* xb = P.x + (size_t)b * (CEEG * TLEN);
    for (int nt = 0; nt < 8; ++nt) {
        int  t  = nt * 16 + m;
        bool tv = (t < TLEN);
        v8f acc = {0.f, 0.f, 0.f, 0.f, 0.f, 0.f, 0.f, 0.f};
#pragma unroll
        for (int ks = 0; ks < 16; ++ks) {
            int c0 = 4 * ks + 2 * hi;      // B layout mirrors A: N=lane%16
            v2f bfr;
            bfr.x = tv ? xb[c0 * TLEN + t] : 0.f;
            bfr.y = tv ? xb[(c0 + 1) * TLEN + t] : 0.f;
            acc = __builtin_amdgcn_wmma_f32_16x16x4_f32(
                false, afr[ks], false, bfr, (short)0, acc, false, false);
        }
#pragma unroll
        for (int r = 0; r < 8; ++r)        // C/D: row r (+8 for hi lanes)
            upad[r + 8 * hi][32 + nt * 16 + m] = acc[r];
    }
    __syncthreads();

    // ====== Stage B: 63-tap temporal conv + BN2 + ELU =======================
    // Each lane owns t = lane + {0,32,64,96}; wtf[f][k] is a uniform broadcast.
    for (int o = 0; o < 16; ++o) {
        const int f = o >> 1;
        const float* __restrict__ urow = &upad[o][0];
        float s0 = 0.f, s1 = 0.f, s2 = 0.f, s3 = 0.f;
#pragma unroll 9
        for (int k = 0; k < 63; ++k) {
            float w = wtf[f][k];
            int j = lane + k + 1;          // = (t + k - 31) + 32
            s0 += w * urow[j];
            s1 += w * urow[j + 32];
            s2 += w * urow[j + 64];
            s3 += w * urow[j + 96];
        }
        float a2 = A2s[o], b2 = B2s[o];
        float z0 = a2 * s0 + b2, z1 = a2 * s1 + b2;
        float z2 = a2 * s2 + b2, z3 = a2 * s3 + b2;
        ebuf[o][lane]      = (z0 > 0.f) ? z0 : (expf(z0) - 1.f);
        ebuf[o][lane + 32] = (z1 > 0.f) ? z1 : (expf(z1) - 1.f);
        ebuf[o][lane + 64] = (z2 > 0.f) ? z2 : (expf(z2) - 1.f);
        if (lane + 96 < TLEN)
            ebuf[o][lane + 96] = (z3 > 0.f) ? z3 : (expf(z3) - 1.f);
    }
    __syncthreads();

    // ====== avg pool (1,5) ==================================================
    for (int idx = lane; idx < 16 * 25; idx += 32) {
        int o = idx / 25, s = idx % 25;
        float acc = 0.f;
#pragma unroll
        for (int j = 0; j < 5; ++j) acc += ebuf[o][5 * s + j];
        pool[o][s] = acc * 0.2f;
    }
    __syncthreads();

    // ====== token projection + cls + positional embedding ===================
    for (int idx = lane; idx < 6 * 16; idx += 32) {
        int tok = idx / 16, d = idx % 16;
        if (tok == 0) {
            zs[0][d] = P.cls_token[d] + P.pe[d];
        } else {
            int i = tok - 1;
            float acc = P.proj_b[d];
#pragma unroll
            for (int r = 0; r < 80; ++r)
                acc += P.proj_w[d * 80 + r] * pool[r / 5][i * 5 + (r % 5)];
            zs[tok][d] = acc + P.pe[tok * 16 + d];
        }
    }
    __syncthreads();

    // ====== transformer: 6 layers, S=6 tokens, D=16, H=8, QK=8 ==============
    const float scale = 0.35355339059327373f;   // 8^-0.5
    for (int li = 0; li < 6; ++li) {
        // LN1
        if (lane < 6) {
            float mean = 0.f;
            for (int d = 0; d < 16; ++d) mean += zs[lane][d];
            mean *= (1.f / 16.f);
            float var = 0.f;
            for (int d = 0; d < 16; ++d) { float df = zs[lane][d] - mean; var += df * df; }
            float rs = rsqrtf(var * (1.f / 16.f) + EPS);
            for (int d = 0; d < 16; ++d)
                h1[lane][d] = (zs[lane][d] - mean) * rs * P.ln1_g[li * 16 + d] + P.ln1_b[li * 16 + d];
        }
        __syncthreads();
        // qkv projection: (6,16)@(16,192)
        for (int idx = lane; idx < 6 * 192; idx += 32) {
            int tok = idx / 192, j = idx % 192;
            qkvs[tok][j] = P.qkv_b[li * 192 + j] +
                           dotN<16>(P.qkv_w + ((size_t)li * 192 + j) * 16, &h1[tok][0]);
        }
        __syncthreads();
        // attention per (head, query-token)
        for (int idx = lane; idx < 48; idx += 32) {
            int h = idx / 6, s = idx % 6;
            float sc[6], mx = -3.4e38f;
            for (int t2 = 0; t2 < 6; ++t2) {
                float dot = 0.f;
#pragma unroll
                for (int d = 0; d < 8; ++d)
                    dot += qkvs[s][h * 8 + d] * qkvs[t2][64 + h * 8 + d];
                sc[t2] = dot * scale;
                mx = fmaxf(mx, sc[t2]);
            }
            float sum = 0.f;
            for (int t2 = 0; t2 < 6; ++t2) { sc[t2] = expf(sc[t2] - mx); sum += sc[t2]; }
            float inv = 1.f / sum;
#pragma unroll
            for (int d = 0; d < 8; ++d) {
                float od = 0.f;
                for (int t2 = 0; t2 < 6; ++t2)
                    od += sc[t2] * qkvs[t2][128 + h * 8 + d];
                attn_o[s][h * 8 + d] = od * inv;
            }
        }
        __syncthreads();
        // out projection + residual
        for (int idx = lane; idx < 6 * 16; idx += 32) {
            int tok = idx / 16, d = idx % 16;
            zs[tok][d] += P.out_b[li * 16 + d] +
                          dotN<64>(P.out_w + ((size_t)li * 16 + d) * 64, &attn_o[tok][0]);
        }
        __syncthreads();
        // LN2
        if (lane < 6) {
            float mean = 0.f;
            for (int d = 0; d < 16; ++d) mean += zs[lane][d];
            mean *= (1.f / 16.f);
            float var = 0.f;
            for (int d = 0; d < 16; ++d) { float df = zs[lane][d] - mean; var += df * df; }
            float rs = rsqrtf(var * (1.f / 16.f) + EPS);
            for (int d = 0; d < 16; ++d)
                h2[lane][d] = (zs[lane][d] - mean) * rs * P.ln2_g[li * 16 + d] + P.ln2_b[li * 16 + d];
        }
        __syncthreads();
        // MLP fc1 + exact GELU
        for (int idx = lane; idx < 6 * 32; idx += 32) {
            int tok = idx / 32, j = idx % 32;
            float acc = P.mlp_b1[li * 32 + j] +
                        dotN<16>(P.mlp_w1 + ((size_t)li * 32 + j) * 16, &h2[tok][0]);
            m1[tok][j] = 0.5f * acc * (1.f + erff(acc * 0.7071067811865475f));
        }
        __syncthreads();
        // MLP fc2 + residual
        for (int idx = lane; idx < 6 * 16; idx += 32) {
            int tok = idx / 16, d = idx % 16;
            zs[tok][d] += P.mlp_b2[li * 16 + d] +
                          dotN<32>(P.mlp_w2 + ((size_t)li * 16 + d) * 32, &m1[tok][0]);
        }
        __syncthreads();
    }

    // ====== head: LN(cls) -> 2-way logits -> softmax ========================
    if (lane == 0) {
        float mean = 0.f;
        for (int d = 0; d < 16; ++d) mean += zs[0][d];
        mean *= (1.f / 16.f);
        float var = 0.f;
        for (int d = 0; d < 16; ++d) { float df = zs[0][d] - mean; var += df * df; }
        float rs = rsqrtf(var * (1.f / 16.f) + EPS);
        float l0 = P.head_b[0], l1 = P.head_b[1];
        for (int d = 0; d < 16; ++d) {
            float hh = (zs[0][d] - mean) * rs * P.head_ln_g[d] + P.head_ln_b[d];
            l0 += P.head_w[d] * hh;
            l1 += P.head_w[16 + d] * hh;
        }
        float mx = fmaxf(l0, l1);
        float e0 = expf(l0 - mx), e1 = expf(l1 - mx);
        float inv = 1.f / (e0 + e1);
        P.out[2 * b]     = e0 * inv;
        P.out[2 * b + 1] = e1 * inv;
    }
}

extern "C" void kernel_launch(void* const* d_in, const int* in_sizes, int n_in,
                              void* d_out, int out_size, void* d_ws, size_t ws_size,
                              hipStream_t stream) {
    (void)in_sizes; (void)n_in; (void)out_size; (void)d_ws; (void)ws_size;
    EegParams p;
    p.x         = (const float*)d_in[0];
    p.conv_t_w  = (const float*)d_in[1];
    p.conv_t_b  = (const float*)d_in[2];
    p.bn1_g     = (const float*)d_in[3];
    p.bn1_b     = (const float*)d_in[4];
    p.bn1_m     = (const float*)d_in[5];
    p.bn1_v     = (const float*)d_in[6];
    p.conv_s_w  = (const float*)d_in[7];
    p.conv_s_b  = (const float*)d_in[8];
    p.bn2_g     = (const float*)d_in[9];
    p.bn2_b     = (const float*)d_in[10];
    p.bn2_m     = (const float*)d_in[11];
    p.bn2_v     = (const float*)d_in[12];
    p.proj_w    = (const float*)d_in[13];
    p.proj_b    = (const float*)d_in[14];
    p.cls_token = (const float*)d_in[15];
    p.pe        = (const float*)d_in[16];
    p.qkv_w     = (const float*)d_in[17];
    p.qkv_b     = (const float*)d_in[18];
    p.out_w     = (const float*)d_in[19];
    p.out_b     = (const float*)d_in[20];
    p.ln1_g     = (const float*)d_in[21];
    p.ln1_b     = (const float*)d_in[22];
    p.ln2_g     = (const float*)d_in[23];
    p.ln2_b     = (const float*)d_in[24];
    p.mlp_w1    = (const float*)d_in[25];
    p.mlp_b1    = (const float*)d_in[26];
    p.mlp_w2    = (const float*)d_in[27];
    p.mlp_b2    = (const float*)d_in[28];
    p.head_ln_g = (const float*)d_in[29];
    p.head_ln_b = (const float*)d_in[30];
    p.head_w    = (const float*)d_in[31];
    p.head_b    = (const float*)d_in[32];
    p.out       = (float*)d_out;

    hipLaunchKernelGGL(eeg_vit_fused, dim3(BATCH), dim3(32), 0, stream, p);
}